// SimCAM_82643760710042
// MI455X (gfx1250) — compile-verified
//
#include <hip/hip_runtime.h>
#include <hip/hip_bf16.h>
#include <math.h>

// SimCAM: decom = x0 (5184x256) @ x1^T (256x5184); global max; relu(decom/max);
// row sums -> map0, col sums -> map1; bilinear resize (2,72,72)->(2,224,224).
// Two-pass tile recompute via V_WMMA_F32_16X16X4_F32; never materializes decom.

typedef float v2f __attribute__((ext_vector_type(2)));
typedef float v8f __attribute__((ext_vector_type(8)));

#define HW       72
#define CH       256
#define MROWS    5184          // 72*72
#define OUT_H    224
#define OUT_W    224

// Order-preserving map float -> unsigned for atomicMax-based float max.
__device__ __forceinline__ unsigned map_f32(float f) {
    unsigned u = __float_as_uint(f);
    return (u & 0x80000000u) ? ~u : (u | 0x80000000u);
}
__device__ __forceinline__ float unmap_f32(unsigned u) {
    unsigned b = (u & 0x80000000u) ? (u ^ 0x80000000u) : ~u;
    return __uint_as_float(b);
}

__global__ void __launch_bounds__(256)
simcam_init_kernel(unsigned* __restrict__ maxslot, float* __restrict__ sums) {
    int idx = blockIdx.x * 256 + threadIdx.x;
    if (idx == 0) *maxslot = 0u;              // below every mapped float
    if (idx < 2 * MROWS) sums[idx] = 0.0f;
}

// Compute one 16x16 tile of A*B^T for this wave. A,B are 5184x256 row-major.
// Returns v8f accumulator in the ISA C/D layout:
//   lane = (half<<4)|r ; acc[v] = tile element (m0 + v + 8*half, n0 + r)
__device__ __forceinline__ v8f simcam_tile(const float* __restrict__ A,
                                           const float* __restrict__ B,
                                           int m0, int n0, int r, int half) {
    // 32-bit A-matrix 16x4 layout: lanes 0-15 hold {K0,K1}, lanes 16-31 {K2,K3},
    // row M = r. B (4x16) is symmetric here because B-tile column n is x1 row n.
    const float* arow = A + (size_t)(m0 + r) * CH + 2 * half;
    const float* brow = B + (size_t)(n0 + r) * CH + 2 * half;
    v8f acc = {};
#pragma unroll 4
    for (int kk = 0; kk < CH; kk += 4) {
        v2f a = *(const v2f*)(arow + kk);
        v2f b = *(const v2f*)(brow + kk);
        // (neg_a, A, neg_b, B, c_mod, C, reuse_a, reuse_b)
        acc = __builtin_amdgcn_wmma_f32_16x16x4_f32(
            false, a, false, b, (short)0, acc, false, false);
    }
    return acc;
}

__global__ void __launch_bounds__(256)
simcam_max_kernel(const float* __restrict__ A, const float* __restrict__ B,
                  unsigned* __restrict__ maxslot) {
    const int tid  = threadIdx.x;
    const int wave = tid >> 5;
    const int lane = tid & 31;
    const int r    = lane & 15;
    const int half = lane >> 4;
    const int m0 = (blockIdx.x * 2 + (wave & 1)) * 16;
    const int n0 = (blockIdx.y * 4 + (wave >> 1)) * 16;

    v8f acc = simcam_tile(A, B, m0, n0, r, half);

    float mx = acc[0];
#pragma unroll
    for (int v = 1; v < 8; ++v) mx = fmaxf(mx, acc[v]);
#pragma unroll
    for (int off = 16; off >= 1; off >>= 1)
        mx = fmaxf(mx, __shfl_xor(mx, off, 32));
    if (lane == 0) atomicMax(maxslot, map_f32(mx));
}

__global__ void __launch_bounds__(256)
simcam_sum_kernel(const float* __restrict__ A, const float* __restrict__ B,
                  const unsigned* __restrict__ maxslot,
                  float* __restrict__ rowsum, float* __restrict__ colsum) {
    const int tid  = threadIdx.x;
    const int wave = tid >> 5;
    const int lane = tid & 31;
    const int r    = lane & 15;
    const int half = lane >> 4;
    const int m0 = (blockIdx.x * 2 + (wave & 1)) * 16;
    const int n0 = (blockIdx.y * 4 + (wave >> 1)) * 16;

    v8f acc = simcam_tile(A, B, m0, n0, r, half);

    const float inv = 1.0f / unmap_f32(*maxslot);

    float rv[8];
    float colp = 0.0f;
#pragma unroll
    for (int v = 0; v < 8; ++v) {
        float s = acc[v] * inv;
        rv[v] = s > 0.0f ? s : 0.0f;
        colp += rv[v];
    }
    // Column n0+r: combine the two half-wave partials (rows 0-7 and 8-15).
    colp += __shfl_xor(colp, 16, 32);
    if (half == 0) atomicAdd(colsum + n0 + r, colp);

    // Row m0+v+8*half: reduce across the 16 lanes of this half (over n).
#pragma unroll
    for (int v = 0; v < 8; ++v) {
#pragma unroll
        for (int off = 1; off < 16; off <<= 1)
            rv[v] += __shfl_xor(rv[v], off, 32);
    }
    if (r == 0) {
#pragma unroll
        for (int v = 0; v < 8; ++v)
            atomicAdd(rowsum + m0 + v + 8 * half, rv[v]);
    }
}

__global__ void __launch_bounds__(256)
simcam_resize_kernel(const float* __restrict__ sums, float* __restrict__ out) {
    int idx = blockIdx.x * 256 + threadIdx.x;
    if (idx >= 2 * OUT_H * OUT_W) return;
    int n   = idx / (OUT_H * OUT_W);
    int rem = idx % (OUT_H * OUT_W);
    int oy  = rem / OUT_W;
    int ox  = rem % OUT_W;

    const float s = (float)HW / (float)OUT_H;   // same scale both dims
    float ys = fmaxf(((float)oy + 0.5f) * s - 0.5f, 0.0f);
    float xs = fmaxf(((float)ox + 0.5f) * s - 0.5f, 0.0f);
    int y0 = (int)floorf(ys), x0 = (int)floorf(xs);
    int y1 = min(y0 + 1, HW - 1), x1 = min(x0 + 1, HW - 1);
    float wy = ys - (float)y0, wx = xs - (float)x0;

    const float* m = sums + (size_t)n * MROWS;  // 72x72 map, row-major
    float a = m[y0 * HW + x0];
    float b = m[y0 * HW + x1];
    float c = m[y1 * HW + x0];
    float d = m[y1 * HW + x1];
    out[idx] = a * (1.0f - wy) * (1.0f - wx) + b * (1.0f - wy) * wx +
               c * wy * (1.0f - wx) + d * wy * wx;
}

extern "C" void kernel_launch(void* const* d_in, const int* in_sizes, int n_in,
                              void* d_out, int out_size, void* d_ws, size_t ws_size,
                              hipStream_t stream) {
    (void)in_sizes; (void)n_in; (void)out_size; (void)ws_size;
    const float* x  = (const float*)d_in[0];     // (2, 72, 72, 256) fp32
    const float* A  = x;                          // x[0] as 5184x256
    const float* Bm = x + (size_t)MROWS * CH;     // x[1] as 5184x256
    // d_in[1], d_in[2] are H=224, W=224 scalars (hardcoded).

    unsigned* maxslot = (unsigned*)d_ws;
    float*    sums    = (float*)((char*)d_ws + 64); // [0..5183]=rows, [5184..]=cols

    simcam_init_kernel<<<(2 * MROWS + 255) / 256, 256, 0, stream>>>(maxslot, sums);

    dim3 grid(MROWS / 32, MROWS / 64);            // 162 x 81, waves tile 2x4
    simcam_max_kernel<<<grid, 256, 0, stream>>>(A, Bm, maxslot);
    simcam_sum_kernel<<<grid, 256, 0, stream>>>(A, Bm, maxslot,
                                                sums, sums + MROWS);

    simcam_resize_kernel<<<(2 * OUT_H * OUT_W + 255) / 256, 256, 0, stream>>>(
        sums, (float*)d_out);
}